// MixhopDecoder_51616916963866
// MI455X (gfx1250) — compile-verified
//
#include <hip/hip_runtime.h>
#include <hip/hip_bf16.h>

// ---------------- MixHop GCN decoder for gfx1250 (MI455X) ----------------
// f32 end-to-end; dense GEMMs via V_WMMA_F32_16X16X4_F32 (exact f32 math),
// weights staged in LDS via GLOBAL_LOAD_ASYNC_TO_LDS_B128 (ASYNCcnt),
// sparse propagation via global f32 atomics (L2-resident).

typedef float v2f __attribute__((ext_vector_type(2)));
typedef float v8f __attribute__((ext_vector_type(8)));

#define FDIM 192
#define BN_EPS 1e-5f
#define SLOPE 0.1f
#define NODE_NUM 1000
#define N_GRAPH 32

// ---------- small utility kernels ----------
__global__ void mh_fill(float* __restrict__ p, float v, int n) {
    int i = blockIdx.x * blockDim.x + threadIdx.x;
    if (i < n) p[i] = v;
}

__global__ void mh_deg_accum(float* __restrict__ deg, const long long* __restrict__ col, int E) {
    int e = blockIdx.x * blockDim.x + threadIdx.x;
    if (e < E) atomicAdd(&deg[col[e]], 1.0f);
}

__global__ void mh_deg_to_dis(float* __restrict__ deg, int n) {
    int i = blockIdx.x * blockDim.x + threadIdx.x;
    if (i < n) deg[i] = rsqrtf(deg[i]);   // deg >= 1 always (self loops)
}

__global__ void mh_compute_norm(float* __restrict__ norm, const float* __restrict__ dis,
                                const long long* __restrict__ row, const long long* __restrict__ col,
                                int E) {
    int e = blockIdx.x * blockDim.x + threadIdx.x;
    if (e < E) norm[e] = dis[row[e]] * dis[col[e]];
}

// out[i,f] = dis[i]^2 * in[i,f]   (self-loop term, also initializes buffer)
__global__ void mh_init_selfloop(float* __restrict__ out, const float* __restrict__ in,
                                 const float* __restrict__ dis, int d, int total) {
    int idx = blockIdx.x * blockDim.x + threadIdx.x;
    if (idx >= total) return;
    int i = idx / d;
    float s = dis[i];
    out[idx] = s * s * in[idx];
}

// out[col[e], :] += norm[e] * in[row[e], :]
__global__ void mh_scatter(float* __restrict__ out, const float* __restrict__ in,
                           const long long* __restrict__ row, const long long* __restrict__ col,
                           const float* __restrict__ norm, int d, int E) {
    int e = blockIdx.x * blockDim.x + threadIdx.x;
    if (e >= E) return;
    long long rs = row[e], cs = col[e];
    float w = norm[e];
    const float4* __restrict__ src = (const float4*)(in + rs * (long long)d);
    float* __restrict__ dst = out + cs * (long long)d;
    int nc = d >> 2;
    for (int c = 0; c < nc; ++c) {
        float4 v = src[c];
        atomicAdd(dst + 4 * c + 0, w * v.x);
        atomicAdd(dst + 4 * c + 1, w * v.y);
        atomicAdd(dst + 4 * c + 2, w * v.z);
        atomicAdd(dst + 4 * c + 3, w * v.w);
    }
}

// ---------- WMMA GEMM:  O[:, colOff:colOff+64] = H[N x d] @ W[d x 64] + bias ----------
// One wave computes a 16x64 tile (4 accumulators) with V_WMMA_F32_16X16X4_F32.
// A(16x4): lane&15 = M, lane>>4 picks K pair {0,1}/{2,3} (VGPR0=Klo, VGPR1=Khi)
// B(4x16): lane&15 = N, same K grouping.  D: VGPR v -> M = v + 8*(lane>>4).
// W (d x 64, <=48KB) is staged into LDS once per block with the CDNA5
// async-copy path (GLOBAL_LOAD_ASYNC_TO_LDS_B128 + s_wait_asynccnt).
typedef __attribute__((address_space(3))) float lds_float;

__global__ void __launch_bounds__(256)
mh_gemm_wmma(const float* __restrict__ H, int d,
             const float* __restrict__ W,     // d x 64, row-major
             const float* __restrict__ bias,  // 64
             float* __restrict__ O,           // N x 192
             int colOff, int mtiles) {
    __shared__ float sW[FDIM * 64];           // 48 KB max, row-major copy of W

    // --- cooperative async copy global -> LDS (16B per lane per iteration) ---
    {
        unsigned ldsBase = (unsigned)(uintptr_t)(lds_float*)&sW[0];
        unsigned long long gBase = (unsigned long long)(uintptr_t)W;
        int nbytes = d * 64 * 4;              // multiple of blockDim*16 for d=32/192
        for (int off = threadIdx.x * 16; off < nbytes; off += blockDim.x * 16) {
            unsigned lds = ldsBase + (unsigned)off;
            unsigned long long ga = gBase + (unsigned long long)off;
            asm volatile("global_load_async_to_lds_b128 %0, %1, off"
                         :: "v"(lds), "v"(ga) : "memory");
        }
        asm volatile("s_wait_asynccnt 0x0" ::: "memory");
    }
    __syncthreads();

    int wave = threadIdx.x >> 5;
    int lane = threadIdx.x & 31;
    int mt = blockIdx.x * 8 + wave;           // one 16-row strip per wave
    if (mt >= mtiles) return;                 // uniform per wave
    int m0 = mt << 4;
    int r  = lane & 15;                       // M for A, N for B
    int kg = (lane >> 4) << 1;                // K-pair base: 0 or 2

    v8f acc0 = {}, acc1 = {}, acc2 = {}, acc3 = {};
    const float* __restrict__ Arow = H + (long long)(m0 + r) * d;

    for (int k0 = 0; k0 < d; k0 += 4) {
        v2f a;
        a.x = Arow[k0 + kg];
        a.y = Arow[k0 + kg + 1];
        const float* wlo = &sW[(k0 + kg) * 64 + r];
        const float* whi = wlo + 64;
        v2f b0, b1, b2, b3;
        b0.x = wlo[0];  b0.y = whi[0];
        b1.x = wlo[16]; b1.y = whi[16];
        b2.x = wlo[32]; b2.y = whi[32];
        b3.x = wlo[48]; b3.y = whi[48];
        acc0 = __builtin_amdgcn_wmma_f32_16x16x4_f32(false, a, false, b0, (short)0, acc0, false, false);
        acc1 = __builtin_amdgcn_wmma_f32_16x16x4_f32(false, a, false, b1, (short)0, acc1, false, false);
        acc2 = __builtin_amdgcn_wmma_f32_16x16x4_f32(false, a, false, b2, (short)0, acc2, false, false);
        acc3 = __builtin_amdgcn_wmma_f32_16x16x4_f32(false, a, false, b3, (short)0, acc3, false, false);
    }

    int half = lane >> 4;
    float bc0 = bias[r], bc1 = bias[16 + r], bc2 = bias[32 + r], bc3 = bias[48 + r];
    float* Obase = O + (long long)m0 * FDIM + colOff + r;
#pragma unroll
    for (int v = 0; v < 8; ++v) {
        float* po = Obase + (long long)(v + 8 * half) * FDIM;
        po[0]  = acc0[v] + bc0;
        po[16] = acc1[v] + bc1;
        po[32] = acc2[v] + bc2;
        po[48] = acc3[v] + bc3;
    }
}

// ---------- BatchNorm (training stats) + LeakyReLU ----------
__global__ void mh_bn_stats(const float* __restrict__ h, float* __restrict__ sums,
                            float* __restrict__ sqs, int total) {
    __shared__ float ss[FDIM];
    __shared__ float sq[FDIM];
    for (int c = threadIdx.x; c < FDIM; c += blockDim.x) { ss[c] = 0.f; sq[c] = 0.f; }
    __syncthreads();
    for (int idx = blockIdx.x * blockDim.x + threadIdx.x; idx < total;
         idx += gridDim.x * blockDim.x) {
        float v = h[idx];
        int c = idx % FDIM;
        atomicAdd(&ss[c], v);
        atomicAdd(&sq[c], v * v);
    }
    __syncthreads();
    for (int c = threadIdx.x; c < FDIM; c += blockDim.x) {
        atomicAdd(&sums[c], ss[c]);
        atomicAdd(&sqs[c], sq[c]);
    }
}

__global__ void mh_bn_finalize(const float* __restrict__ sums, const float* __restrict__ sqs,
                               const float* __restrict__ gamma, const float* __restrict__ beta,
                               float* __restrict__ params, float invn) {
    int c = threadIdx.x;
    if (c >= FDIM) return;
    float mean = sums[c] * invn;
    float var  = sqs[c] * invn - mean * mean;
    float inv  = rsqrtf(var + BN_EPS);
    float sc   = gamma[c] * inv;
    params[c]         = sc;                  // scale
    params[FDIM + c]  = beta[c] - mean * sc; // shift
}

__global__ void mh_bn_apply_leaky(float* __restrict__ h, const float* __restrict__ params,
                                  int total) {
    int idx = blockIdx.x * blockDim.x + threadIdx.x;
    if (idx >= total) return;
    int c = idx % FDIM;
    float v = params[c] * h[idx] + params[FDIM + c];
    h[idx] = (v >= 0.f) ? v : SLOPE * v;
}

// ---------- Power-mean pooling over nodes ----------
__global__ void mh_pool(const float* __restrict__ h, float* __restrict__ pooled,
                        const float* __restrict__ p_ptr) {
    int g = blockIdx.x;          // graph
    int c = threadIdx.x;         // channel (192 threads)
    float p = p_ptr[0];
    const float* __restrict__ base = h + (long long)g * NODE_NUM * FDIM + c;
    float s = 0.f;
    for (int i = 0; i < NODE_NUM; ++i) {
        float v = base[(long long)i * FDIM];
        v = fminf(fmaxf(v, 1e-7f), 100.0f);
        s += powf(v, p);
    }
    float m = s * (1.0f / NODE_NUM);
    m = fminf(fmaxf(m, 1e-7f), 100.0f);
    pooled[g * FDIM + c] = powf(m, 1.0f / p);
}

// ---------- Head: logits + argmax ----------
__global__ void mh_head(const float* __restrict__ pooled, const float* __restrict__ wp,
                        const float* __restrict__ bp, float* __restrict__ out) {
    int g = threadIdx.x;
    if (g >= N_GRAPH) return;
    float o0 = bp[0], o1 = bp[1];
    const float* pv = pooled + g * FDIM;
    for (int c = 0; c < FDIM; ++c) {
        float v = pv[c];
        o0 += v * wp[c * 2 + 0];
        o1 += v * wp[c * 2 + 1];
    }
    out[g * 2 + 0] = o0;
    out[g * 2 + 1] = o1;
    out[N_GRAPH * 2 + g] = (o1 > o0) ? 1.0f : 0.0f;   // argmax (ties -> 0)
}

// ---------------- host-side orchestration ----------------
extern "C" void kernel_launch(void* const* d_in, const int* in_sizes, int n_in,
                              void* d_out, int out_size, void* d_ws, size_t ws_size,
                              hipStream_t stream) {
    const float*      x    = (const float*)d_in[0];
    const long long*  ei   = (const long long*)d_in[1];
    const float*      w1   = (const float*)d_in[2];
    const float*      b1   = (const float*)d_in[3];
    const float*      w2   = (const float*)d_in[4];
    const float*      b2   = (const float*)d_in[5];
    const float*      w3   = (const float*)d_in[6];
    const float*      b3   = (const float*)d_in[7];
    const float*      g1   = (const float*)d_in[8];
    const float*      be1  = (const float*)d_in[9];
    const float*      g2   = (const float*)d_in[10];
    const float*      be2  = (const float*)d_in[11];
    const float*      g3   = (const float*)d_in[12];
    const float*      be3  = (const float*)d_in[13];
    const float*      p    = (const float*)d_in[14];
    const float*      wp   = (const float*)d_in[15];
    const float*      bp   = (const float*)d_in[16];
    float*            out  = (float*)d_out;

    const int N = in_sizes[0] / 32;      // 32000
    const int E = in_sizes[1] / 2;       // 1024000
    const long long* row = ei;
    const long long* col = ei + E;

    // workspace layout (floats)
    float* ws = (float*)d_ws;
    size_t off = 0;
    float* norm = ws + off; off += (size_t)E;
    float* dis  = ws + off; off += (size_t)N;
    float* h1   = ws + off; off += (size_t)N * FDIM;
    float* h2   = ws + off; off += (size_t)N * FDIM;
    float* hA   = ws + off; off += (size_t)N * FDIM;
    float* hB   = ws + off; off += (size_t)N * FDIM;
    float* stats  = ws + off; off += 2 * FDIM;   // sums | sumsq
    float* params = ws + off; off += 2 * FDIM;   // scale | shift
    float* pooled = ws + off; off += (size_t)N_GRAPH * FDIM;

    const int T = 256;
    auto blks = [](long long n, int t) { return (unsigned)((n + t - 1) / t); };

    // ---- gcn_norm ----
    mh_fill<<<blks(N, T), T, 0, stream>>>(dis, 1.0f, N);                 // deg starts at 1 (self loop)
    mh_deg_accum<<<blks(E, T), T, 0, stream>>>(dis, col, E);
    mh_deg_to_dis<<<blks(N, T), T, 0, stream>>>(dis, N);
    mh_compute_norm<<<blks(E, T), T, 0, stream>>>(norm, dis, row, col, E);

    const int mtiles = N / 16;                          // 2000
    const unsigned gemm_blocks = (unsigned)((mtiles + 7) / 8);

    // one MixHop layer: Hout = concat(Hin@W0, (A Hin)@W1, (A A Hin)@W2)+b, then BN+leaky
    auto layer = [&](const float* Hin, int din, const float* W, const float* b,
                     const float* gam, const float* bet, float* Hout) {
        long long tot_in = (long long)N * din;
        // h1 = A_hat @ Hin
        mh_init_selfloop<<<blks(tot_in, T), T, 0, stream>>>(h1, Hin, dis, din, (int)tot_in);
        mh_scatter<<<blks(E, T), T, 0, stream>>>(h1, Hin, row, col, norm, din, E);
        // h2 = A_hat @ h1
        mh_init_selfloop<<<blks(tot_in, T), T, 0, stream>>>(h2, h1, dis, din, (int)tot_in);
        mh_scatter<<<blks(E, T), T, 0, stream>>>(h2, h1, row, col, norm, din, E);
        // three WMMA GEMMs into the 192-wide concat
        mh_gemm_wmma<<<gemm_blocks, 256, 0, stream>>>(Hin, din, W,                        b,       Hout, 0,   mtiles);
        mh_gemm_wmma<<<gemm_blocks, 256, 0, stream>>>(h1,  din, W + (size_t)din * 64,     b + 64,  Hout, 64,  mtiles);
        mh_gemm_wmma<<<gemm_blocks, 256, 0, stream>>>(h2,  din, W + (size_t)2 * din * 64, b + 128, Hout, 128, mtiles);
        // BN (training stats) + LeakyReLU
        long long tot = (long long)N * FDIM;
        mh_fill<<<blks(2 * FDIM, T), T, 0, stream>>>(stats, 0.f, 2 * FDIM);
        mh_bn_stats<<<480, T, 0, stream>>>(Hout, stats, stats + FDIM, (int)tot);
        mh_bn_finalize<<<1, FDIM, 0, stream>>>(stats, stats + FDIM, gam, bet, params, 1.0f / (float)N);
        mh_bn_apply_leaky<<<blks(tot, T), T, 0, stream>>>(Hout, params, (int)tot);
    };

    layer(x,  32,   w1, b1, g1, be1, hA);
    layer(hA, FDIM, w2, b2, g2, be2, hB);
    layer(hB, FDIM, w3, b3, g3, be3, hA);

    // ---- pooling + head ----
    mh_pool<<<N_GRAPH, FDIM, 0, stream>>>(hA, pooled, p);
    mh_head<<<1, 64, 0, stream>>>(pooled, wp, bp, out);
}